// SE3Comp_47622597378731
// MI455X (gfx1250) — compile-verified
//
#include <hip/hip_runtime.h>
#include <math.h>

// SE(3) composition: out[b] = R7( ExpSE3(xi[b]) * SE3(Tg[b]) )
// B = 2e6, memory-bound: 160 MB @ 23.3 TB/s ~= 7 us. Strategy: async-LDS
// staged, fully coalesced non-temporal B128 traffic in and out; scalar VALU
// math per element (WMMA inapplicable: 2M independent 3x3 products share no
// K dimension, so they cannot be batched into 16x16xK tiles productively).

#define TILE 256
#define TG_CHUNKS (TILE * 28 / 16)   // 448 x 16B = TILE*7 floats
#define XI_CHUNKS (TILE * 24 / 16)   // 384 x 16B = TILE*6 floats

__device__ __forceinline__ void async_ld_b128(unsigned la, unsigned long long ga) {
    asm volatile("global_load_async_to_lds_b128 %0, %1, off th:TH_LOAD_NT"
                 :: "v"(la), "v"(ga) : "memory");
}
__device__ __forceinline__ void async_ld_b32(unsigned la, unsigned long long ga) {
    asm volatile("global_load_async_to_lds_b32 %0, %1, off"
                 :: "v"(la), "v"(ga) : "memory");
}
__device__ __forceinline__ void async_st_b128(unsigned long long ga, unsigned la) {
    asm volatile("global_store_async_from_lds_b128 %0, %1, off th:TH_STORE_NT"
                 :: "v"(ga), "v"(la) : "memory");
}
__device__ __forceinline__ void async_st_b32(unsigned long long ga, unsigned la) {
    asm volatile("global_store_async_from_lds_b32 %0, %1, off"
                 :: "v"(ga), "v"(la) : "memory");
}

__global__ __launch_bounds__(TILE) void se3comp_kernel(
    const float* __restrict__ Tg, const float* __restrict__ xi,
    float* __restrict__ out, int n)
{
    // Per-element: 7 floats Tg | 6 floats xi | 7 floats out
    __shared__ __align__(16) float lds[TILE * 20];
    float* ltg  = lds;              // TILE*7 floats  (7168 B, 16B-aligned)
    float* lxi  = lds + TILE * 7;   // TILE*6 floats  (offset 7168, 16B-aligned)
    float* lout = lds + TILE * 13;  // TILE*7 floats  (offset 13312, 16B-aligned)

    const int tid = (int)threadIdx.x;
    const long long tileStart = (long long)blockIdx.x * TILE;
    int elems = n - (int)tileStart;
    if (elems <= 0) return;
    if (elems > TILE) elems = TILE;

    // Low 32 bits of a generic LDS pointer == LDS byte address (flat aperture).
    const unsigned ltgOff  = (unsigned)(size_t)(void*)ltg;
    const unsigned lxiOff  = (unsigned)(size_t)(void*)lxi;
    const unsigned loutOff = (unsigned)(size_t)(void*)lout;

    const unsigned long long tgSrc = (unsigned long long)(const void*)(Tg + tileStart * 7);
    const unsigned long long xiSrc = (unsigned long long)(const void*)(xi + tileStart * 6);
    const unsigned long long oDst  = (unsigned long long)(void*)(out + tileStart * 7);

    // ---- Async copy-in: global -> LDS (ASYNCcnt), 16 B per lane ----
    if (elems == TILE) {
        // Full tile (uniform branch): compile-time chunk counts, unrolled.
        async_ld_b128(ltgOff + (unsigned)tid * 16u, tgSrc + (unsigned long long)tid * 16u);
        if (tid < TG_CHUNKS - TILE) {  // 192 lanes
            unsigned j = (unsigned)tid + TILE;
            async_ld_b128(ltgOff + j * 16u, tgSrc + (unsigned long long)j * 16u);
        }
        async_ld_b128(lxiOff + (unsigned)tid * 16u, xiSrc + (unsigned long long)tid * 16u);
        if (tid < XI_CHUNKS - TILE) {  // 128 lanes
            unsigned j = (unsigned)tid + TILE;
            async_ld_b128(lxiOff + j * 16u, xiSrc + (unsigned long long)j * 16u);
        }
    } else {
        // Partial tail tile: generic b128 loops + b32 remainder.
        const unsigned tgFloats = (unsigned)elems * 7u;
        const unsigned xiFloats = (unsigned)elems * 6u;
        const unsigned tgChunks = (tgFloats * 4u) >> 4;
        const unsigned xiChunks = (xiFloats * 4u) >> 4;
        for (unsigned j = (unsigned)tid; j < tgChunks; j += TILE)
            async_ld_b128(ltgOff + j * 16u, tgSrc + (unsigned long long)j * 16u);
        for (unsigned j = tgChunks * 4u + (unsigned)tid; j < tgFloats; j += TILE)
            async_ld_b32(ltgOff + j * 4u, tgSrc + (unsigned long long)j * 4u);
        for (unsigned j = (unsigned)tid; j < xiChunks; j += TILE)
            async_ld_b128(lxiOff + j * 16u, xiSrc + (unsigned long long)j * 16u);
        for (unsigned j = xiChunks * 4u + (unsigned)tid; j < xiFloats; j += TILE)
            async_ld_b32(lxiOff + j * 4u, xiSrc + (unsigned long long)j * 4u);
    }

    asm volatile("s_wait_asynccnt 0x0" ::: "memory"); // my wave's LDS writes landed
    __syncthreads();                                  // everyone's landed

    if (tid < elems) {
        const float* e  = ltg + tid * 7;   // stride 7: gcd(7,64)=1, conflict-free
        const float* x6 = lxi + tid * 6;
        float tgx = e[0], tgy = e[1], tgz = e[2];
        float qw  = e[3], qx  = e[4], qy  = e[5], qz = e[6];
        float rx = x6[0], ry = x6[1], rz = x6[2];
        float ox = x6[3], oy = x6[4], oz = x6[5];

        // so(3) exp coefficients (match reference small-angle constants exactly)
        float ts   = ox*ox + oy*oy + oz*oz;
        bool  safe = ts > 1e-12f;
        float tss  = safe ? ts : 1.0f;
        float theta = sqrtf(tss);
        float st = sinf(theta), ct = cosf(theta);
        float A  = safe ? (st / theta)                   : 1.0f;
        float Bc = safe ? ((1.0f - ct) / tss)            : 0.0f;
        float Cc = safe ? ((theta - st) / (tss * theta)) : (1.0f / 6.0f);

        float k201 = ox*oy, k202 = ox*oz, k212 = oy*oz;
        float k200 = -(oy*oy + oz*oz);
        float k211 = -(ox*ox + oz*oz);
        float k222 = -(ox*ox + oy*oy);

        float R00 = 1.0f + Bc*k200, R01 = Bc*k201 - A*oz, R02 = Bc*k202 + A*oy;
        float R10 = Bc*k201 + A*oz, R11 = 1.0f + Bc*k211, R12 = Bc*k212 - A*ox;
        float R20 = Bc*k202 - A*oy, R21 = Bc*k212 + A*ox, R22 = 1.0f + Bc*k222;

        float V00 = 1.0f + Cc*k200, V01 = Cc*k201 - Bc*oz, V02 = Cc*k202 + Bc*oy;
        float V10 = Cc*k201 + Bc*oz, V11 = 1.0f + Cc*k211, V12 = Cc*k212 - Bc*ox;
        float V20 = Cc*k202 - Bc*oy, V21 = Cc*k212 + Bc*ox, V22 = 1.0f + Cc*k222;

        float tx = V00*rx + V01*ry + V02*rz;
        float ty = V10*rx + V11*ry + V12*rz;
        float tz = V20*rx + V21*ry + V22*rz;

        // quaternion (w,x,y,z) -> rotation matrix
        float g00 = 1.0f - 2.0f*(qy*qy + qz*qz);
        float g01 = 2.0f*(qx*qy - qz*qw);
        float g02 = 2.0f*(qx*qz + qy*qw);
        float g10 = 2.0f*(qx*qy + qz*qw);
        float g11 = 1.0f - 2.0f*(qx*qx + qz*qz);
        float g12 = 2.0f*(qy*qz - qx*qw);
        float g20 = 2.0f*(qx*qz - qy*qw);
        float g21 = 2.0f*(qy*qz + qx*qw);
        float g22 = 1.0f - 2.0f*(qx*qx + qy*qy);

        // T = Txi * Tg : rotation part and translation part
        float m00 = R00*g00 + R01*g10 + R02*g20;
        float m01 = R00*g01 + R01*g11 + R02*g21;
        float m02 = R00*g02 + R01*g12 + R02*g22;
        float m10 = R10*g00 + R11*g10 + R12*g20;
        float m11 = R10*g01 + R11*g11 + R12*g21;
        float m12 = R10*g02 + R11*g12 + R12*g22;
        float m20 = R20*g00 + R21*g10 + R22*g20;
        float m21 = R20*g01 + R21*g11 + R22*g21;
        float m22 = R20*g02 + R21*g12 + R22*g22;

        float ttx = R00*tgx + R01*tgy + R02*tgz + tx;
        float tty = R10*tgx + R11*tgy + R12*tgz + ty;
        float ttz = R20*tgx + R21*tgy + R22*tgz + tz;

        // matrix -> quaternion, nested-select order identical to reference
        bool c_m22  = m22 < 0.0f;
        bool c_0g1  = m00 > m11;
        bool c_0ln1 = m00 < -m11;
        bool s1 = c_m22 && c_0g1;
        bool s2 = c_m22 && !c_0g1;
        bool s3 = !c_m22 && c_0ln1;

        float t1 = 1.0f + m00 - m11 - m22;
        float t2 = 1.0f - m00 + m11 - m22;
        float t3 = 1.0f - m00 - m11 + m22;
        float t4 = 1.0f + m00 + m11 + m22;

        float a2112 = m21 - m12, a0220 = m02 - m20, a1001 = m10 - m01;
        float p0110 = m01 + m10, p2002 = m20 + m02, p1221 = m12 + m21;

        float tsel = s1 ? t1 : (s2 ? t2 : (s3 ? t3 : t4));
        float q0 = s1 ? a2112 : (s2 ? a0220 : (s3 ? a1001 : t4));
        float q1 = s1 ? t1    : (s2 ? p0110 : (s3 ? p2002 : a2112));
        float q2 = s1 ? p0110 : (s2 ? t2    : (s3 ? p1221 : a0220));
        float q3 = s1 ? p2002 : (s2 ? p1221 : (s3 ? t3    : a1001));

        float scl = 0.5f / sqrtf(tsel);
        q0 *= scl; q1 *= scl; q2 *= scl; q3 *= scl;
        if (q0 < 0.0f) { q0 = -q0; q1 = -q1; q2 = -q2; q3 = -q3; }

        float* o = lout + tid * 7;
        o[0] = ttx; o[1] = tty; o[2] = ttz;
        o[3] = q0;  o[4] = q1;  o[5] = q2;  o[6] = q3;
    }
    __syncthreads();  // DS writes complete before async engine reads LDS

    // ---- Async copy-out: LDS -> global (no VGPR round-trip) ----
    if (elems == TILE) {
        async_st_b128(oDst + (unsigned long long)tid * 16u, loutOff + (unsigned)tid * 16u);
        if (tid < TG_CHUNKS - TILE) {  // 192 lanes
            unsigned j = (unsigned)tid + TILE;
            async_st_b128(oDst + (unsigned long long)j * 16u, loutOff + j * 16u);
        }
    } else {
        const unsigned tgFloats = (unsigned)elems * 7u;
        const unsigned tgChunks = (tgFloats * 4u) >> 4;
        for (unsigned j = (unsigned)tid; j < tgChunks; j += TILE)
            async_st_b128(oDst + (unsigned long long)j * 16u, loutOff + j * 16u);
        for (unsigned j = tgChunks * 4u + (unsigned)tid; j < tgFloats; j += TILE)
            async_st_b32(oDst + (unsigned long long)j * 4u, loutOff + j * 4u);
    }
    // S_ENDPGM performs an implicit wait-idle (drains ASYNCcnt) before retiring.
}

extern "C" void kernel_launch(void* const* d_in, const int* in_sizes, int n_in,
                              void* d_out, int out_size, void* d_ws, size_t ws_size,
                              hipStream_t stream) {
    (void)n_in; (void)out_size; (void)d_ws; (void)ws_size;
    const float* Tg = (const float*)d_in[0];  // (B,7,1) f32
    const float* xi = (const float*)d_in[1];  // (B,6,1) f32
    float* out = (float*)d_out;               // (B,7,1) f32
    int n = in_sizes[0] / 7;
    int blocks = (n + TILE - 1) / TILE;
    se3comp_kernel<<<blocks, TILE, 0, stream>>>(Tg, xi, out, n);
}